// FaceEdgeVertexGCN_75187697484407
// MI455X (gfx1250) — compile-verified
//
#include <hip/hip_runtime.h>

#define NNODES 262144
#define NEDGES 2000000
#define WF 32

typedef __attribute__((ext_vector_type(2))) float v2f;
typedef __attribute__((ext_vector_type(8))) float v8f;

__device__ __forceinline__ float lrelu(float x) { return x > 0.f ? x : 0.01f * x; }

// Monotone order-preserving float<->uint mapping so unsigned atomicMax == float max.
__device__ __forceinline__ unsigned enc_mono(float f) {
    unsigned u = __float_as_uint(f);
    return (u & 0x80000000u) ? ~u : (u | 0x80000000u);
}
__device__ __forceinline__ float dec_mono_or_zero(unsigned u) {
    unsigned b = (u & 0x80000000u) ? (u ^ 0x80000000u) : ~u;
    // reference zeroes non-finite (empty-segment) maxes
    if ((b & 0x7F800000u) == 0x7F800000u) return 0.f;
    return __uint_as_float(b);
}

// enc(-inf) = ~0xFF800000 = 0x007FFFFF : the "empty segment" init value
#define ENC_NEG_INF 0x007FFFFFu

// ---- input LinearBlock: out[n,32] = leaky_relu(x[n,cin] @ w[cin,32] + b) ----
__global__ void lin_kernel(const float* __restrict__ x, const float* __restrict__ w,
                           const float* __restrict__ b, float* __restrict__ out, int cin) {
    int idx = blockIdx.x * blockDim.x + threadIdx.x;
    if (idx >= NNODES * WF) return;
    int node = idx >> 5, j = idx & 31;
    float acc = b[j];
#pragma unroll 6
    for (int k = 0; k < cin; ++k) acc += x[node * cin + k] * w[k * WF + j];
    out[idx] = lrelu(acc);
}

// ---- fill encoded-max buffer with enc(-inf) ----
__global__ void init_maxes(unsigned* __restrict__ m) {
    int idx = blockIdx.x * blockDim.x + threadIdx.x;
    if (idx < NNODES * WF) m[idx] = ENC_NEG_INF;
}

// ---- one wave32 per edge, lane = feature: coalesced 128B gathers + 1 native atomic ----
__global__ void scatter_max(const float* __restrict__ xs, const float* __restrict__ xd,
                            const int* __restrict__ e, unsigned* __restrict__ m) {
    long gid = (long)blockIdx.x * blockDim.x + threadIdx.x;
    long edge = gid >> 5;
    int lane = gid & 31;
    if (edge >= NEDGES) return;
    int s = e[edge];           // e[0][edge]
    int d = e[NEDGES + edge];  // e[1][edge]
    float diff = xd[(long)d * WF + lane] - xs[(long)s * WF + lane];
    atomicMax(&m[(long)d * WF + lane], enc_mono(diff));
}

// ---- per-16-row-tile MLP via V_WMMA_F32_16X16X4_F32 ----
// xd[tile] = xd[tile] + lrelu( cat(xd[tile], maxes[tile])[16,64] @ w[64,32] + b )
__global__ void mlp_wmma(float* __restrict__ xd, const unsigned* __restrict__ m,
                         const float* __restrict__ w, const float* __restrict__ b) {
    int wave = (int)((blockIdx.x * blockDim.x + threadIdx.x) >> 5);
    int lane = threadIdx.x & 31;
    int r16  = lane & 15;   // row within tile (A), col within half (B/C/D)
    int hi   = lane >> 4;   // lane-half selects K pair (A/B) and M offset (C/D)
    long arow = (long)wave * 16 + r16;
    int col = r16;

    v8f c0 = {};  // output cols 0..15
    v8f c1 = {};  // output cols 16..31

#pragma unroll
    for (int i = 0; i < 16; ++i) {
        int kb = i * 4 + hi * 2;  // this lane's K pair: kb, kb+1
        v2f a, b0, b1;
        if (kb < 32) {  // first half of cat(): x_dst
            a.x = xd[arow * WF + kb];
            a.y = xd[arow * WF + kb + 1];
        } else {        // second half: decoded segment maxes (empty -> 0)
            a.x = dec_mono_or_zero(m[arow * WF + kb - 32]);
            a.y = dec_mono_or_zero(m[arow * WF + kb - 31]);
        }
        b0.x = w[kb * WF + col];
        b0.y = w[(kb + 1) * WF + col];
        b1.x = w[kb * WF + 16 + col];
        b1.y = w[(kb + 1) * WF + 16 + col];
        c0 = __builtin_amdgcn_wmma_f32_16x16x4_f32(false, a, false, b0, (short)0, c0, false, false);
        c1 = __builtin_amdgcn_wmma_f32_16x16x4_f32(false, a, false, b1, (short)0, c1, false, false);
    }

    float bc0 = b[col], bc1 = b[16 + col];
#pragma unroll
    for (int r = 0; r < 8; ++r) {
        int mrow = hi ? (r + 8) : r;          // C/D VGPR r holds M=r (lanes 0-15) / M=r+8
        long orow = (long)wave * 16 + mrow;
        float h0 = lrelu(c0[r] + bc0);
        float h1 = lrelu(c1[r] + bc1);
        xd[orow * WF + col]      += h0;       // residual, tile-local -> race-free
        xd[orow * WF + 16 + col] += h1;
    }
}

extern "C" void kernel_launch(void* const* d_in, const int* in_sizes, int n_in,
                              void* d_out, int out_size, void* d_ws, size_t ws_size,
                              hipStream_t stream) {
    const float* x_f = (const float*)d_in[0];
    const float* x_e = (const float*)d_in[1];
    const float* x_v = (const float*)d_in[2];
    const int* e_fe = (const int*)d_in[3];
    const int* e_ev = (const int*)d_in[4];
    const int* e_ef = (const int*)d_in[5];
    const int* e_ve = (const int*)d_in[6];
    const float* wf = (const float*)d_in[7];
    const float* bf = (const float*)d_in[8];
    const float* we = (const float*)d_in[9];
    const float* be = (const float*)d_in[10];
    const float* wv = (const float*)d_in[11];
    const float* bv = (const float*)d_in[12];
    const float* w_f2e = (const float*)d_in[13];
    const float* b_f2e = (const float*)d_in[14];
    const float* w_e2v = (const float*)d_in[15];
    const float* b_e2v = (const float*)d_in[16];

    // outputs live in d_out, concatenated in return order (x_f, x_e, x_v)
    float* xf = (float*)d_out;
    float* xe = xf + (size_t)NNODES * WF;
    float* xv = xe + (size_t)NNODES * WF;
    unsigned* maxes = (unsigned*)d_ws;  // 32 MB scratch

    const int nodeElems = NNODES * WF;             // 8388608
    const int nodeBlks = nodeElems / 256;          // 32768
    const long edgeThreads = (long)NEDGES * 32;    // 64000000
    const int edgeBlks = (int)((edgeThreads + 255) / 256);
    const int mlpBlks = (NNODES / 16) / 8;         // 8 waves/block, 1 tile/wave

    // input linear blocks
    lin_kernel<<<nodeBlks, 256, 0, stream>>>(x_f, wf, bf, xf, 4);
    lin_kernel<<<nodeBlks, 256, 0, stream>>>(x_e, we, be, xe, 6);
    lin_kernel<<<nodeBlks, 256, 0, stream>>>(x_v, wv, bv, xv, 3);

    auto conv = [&](const float* xsrc, float* xdst, const int* e,
                    const float* wm, const float* bm) {
        init_maxes<<<nodeBlks, 256, 0, stream>>>(maxes);
        scatter_max<<<edgeBlks, 256, 0, stream>>>(xsrc, xdst, e, maxes);
        mlp_wmma<<<mlpBlks, 256, 0, stream>>>(xdst, maxes, wm, bm);
    };

    // NB: reference reuses F2E weights for three of the four convs
    conv(xf, xe, e_fe, w_f2e, b_f2e);
    conv(xe, xv, e_ev, w_e2v, b_e2v);
    conv(xe, xf, e_ef, w_f2e, b_f2e);
    conv(xv, xe, e_ve, w_f2e, b_f2e);
}